// GACLayer_51092930953557
// MI455X (gfx1250) — compile-verified
//
#include <hip/hip_runtime.h>
#include <math.h>

#define BB 8
#define NN 8192
#define SS 2048
#define KK 32
#define DD 64
#define BS (BB*SS)            /* 16384 queries */
#define R2 0.01f
#define TH_ 0.5f
#define NEGS 0.2f
#define EPSF 1e-5f

#define W1 4                  /* waves per block in kernel 1 */
#define TILE 1024             /* xyz points staged per LDS tile */
#define FSTRIDE 65            /* padded feature row stride (floats) -> conflict-free */

typedef float v2f __attribute__((ext_vector_type(2)));
typedef float v8f __attribute__((ext_vector_type(8)));
typedef int   v4i_b __attribute__((vector_size(16)));   // matches builtin's V4i param

__device__ __forceinline__ float wave_max32(float v) {
  for (int o = 16; o > 0; o >>= 1) v = fmaxf(v, __shfl_xor(v, o, 32));
  return v;
}
__device__ __forceinline__ float wave_sum32(float v) {
  for (int o = 16; o > 0; o >>= 1) v += __shfl_xor(v, o, 32);
  return v;
}

// ---------------------------------------------------------------------------
// Kernel 1: one wave per query. Ball query (first K in index order) via
// ballot/popc compaction over LDS-staged xyz tiles (staged with gfx1250
// async global->LDS b128 copies when available); shuffle softmax; masked
// mean/max/std aggregation. Writes m_base (BS x 192) and dlog (BS).
// ---------------------------------------------------------------------------
__global__ __launch_bounds__(W1*32) void gac_group_att(
    const float* __restrict__ xyz, const float* __restrict__ points,
    const float* __restrict__ W_att, const float* __restrict__ b_att,
    const int* __restrict__ fps_idx, float* __restrict__ m_base,
    float* __restrict__ dlog)
{
  __shared__ __align__(16) float s_xyz[TILE*3];   // 12 KB
  __shared__ float s_feat[W1][KK*FSTRIDE];        // ~32.5 KB
  __shared__ float s_pq[W1][DD];
  __shared__ int   s_ngh[W1][KK];
  __shared__ float s_w[W1][KK];
  __shared__ float s_watt[DD+3];

  const int tid  = threadIdx.x;
  const int wave = tid >> 5, lane = tid & 31;
  const int bs   = blockIdx.x * W1 + wave;   // all W1 queries share batch b
  const int b    = bs / SS;
  const int q    = fps_idx[bs];
  const size_t prow = (size_t)(b*NN + q);

  const float qx = xyz[prow*3+0], qy = xyz[prow*3+1], qz = xyz[prow*3+2];
  const float qs = qx*qx + qy*qy + qz*qz;
  if (tid < DD+3) s_watt[tid] = W_att[tid];
  const float batt = b_att[0];
  s_pq[wave][lane]      = points[prow*DD + lane];
  s_pq[wave][lane + 32] = points[prow*DD + lane + 32];

  // -------- ball query: first K in-radius indices in ascending order -------
  int found = 0;
  for (int tile = 0; tile < NN/TILE; ++tile) {
    __syncthreads();
    const float* src = xyz + (size_t)b*NN*3 + (size_t)tile*TILE*3;
#if __has_builtin(__builtin_amdgcn_global_load_async_to_lds_b128)
    // 12288 B = 768 x b128; 128 threads -> 6 async copies each (ASYNCcnt path)
    for (int i = tid; i < (TILE*3)/4; i += W1*32) {
      __builtin_amdgcn_global_load_async_to_lds_b128(
          (__attribute__((address_space(1))) v4i_b*)(src + (size_t)i*4),
          (__attribute__((address_space(3))) v4i_b*)(s_xyz + (size_t)i*4),
          0, 0);
    }
#if __has_builtin(__builtin_amdgcn_s_wait_asynccnt)
    __builtin_amdgcn_s_wait_asynccnt(0);
#else
    asm volatile("s_wait_asynccnt 0" ::: "memory");
#endif
#else
    for (int i = tid; i < TILE*3; i += W1*32) s_xyz[i] = src[i];
#endif
    __syncthreads();
    if (found >= KK) continue;
    for (int c = 0; c < TILE && found < KK; c += 32) {
      const int j = c + lane;
      const float px = s_xyz[3*j], py = s_xyz[3*j+1], pz = s_xyz[3*j+2];
      const float dot = qx*px + qy*py + qz*pz;
      const float ps  = px*px + py*py + pz*pz;
      const float sqr = qs + ps - 2.0f*dot;      // matches ref -2ab + a^2 + b^2
      const bool in = (sqr <= R2);               // ref keeps sqr <= r^2
      const unsigned msk = (unsigned)__ballot(in);
      if (msk) {
        const int pos = found + __popc(msk & ((1u << lane) - 1u));
        if (in && pos < KK) s_ngh[wave][pos] = tile*TILE + j;
        found += __popc(msk);
      }
    }
  }
  if (found > KK) found = KK;
  __syncthreads();
  if (found == 0) { if (lane == 0) s_ngh[wave][0] = 0; found = 1; }
  __syncthreads();
  {
    const int first = s_ngh[wave][0];
    for (int i = found + lane; i < KK; i += 32) s_ngh[wave][i] = first;
  }
  __syncthreads();

  // -------- per-neighbor (lane == neighbor slot): gather + score -----------
  const int nidx = s_ngh[wave][lane];
  const size_t nrow = (size_t)(b*NN + nidx);
  const float gx = xyz[nrow*3+0] - qx;
  const float gy = xyz[nrow*3+1] - qy;
  const float gz = xyz[nrow*3+2] - qz;
  float z = gx*s_watt[0] + gy*s_watt[1] + gz*s_watt[2] + batt;
  const float4* pf = (const float4*)(points + nrow*DD);
  for (int d4 = 0; d4 < DD/4; ++d4) {
    const float4 f = pf[d4];
    const int d = d4*4;
    z += (f.x - s_pq[wave][d+0]) * s_watt[3+d+0];
    z += (f.y - s_pq[wave][d+1]) * s_watt[3+d+1];
    z += (f.z - s_pq[wave][d+2]) * s_watt[3+d+2];
    z += (f.w - s_pq[wave][d+3]) * s_watt[3+d+3];
    s_feat[wave][lane*FSTRIDE + d+0] = f.x;
    s_feat[wave][lane*FSTRIDE + d+1] = f.y;
    s_feat[wave][lane*FSTRIDE + d+2] = f.z;
    s_feat[wave][lane*FSTRIDE + d+3] = f.w;
  }
  z = (z >= 0.0f) ? z : NEGS * z;              // leaky relu
  const float mx = wave_max32(z);              // stable softmax over 32 slots
  const float e  = expf(z - mx);
  const float se = wave_sum32(e);
  const float att = e / se;
  const float w   = (att < TH_) ? att : 0.0f;  // threshold (strict <)
  const float adj = (w != 0.0f) ? 1.0f : 0.0f;
  const float deg = wave_sum32(adj);
  s_w[wave][lane] = w;
  if (lane == 0) dlog[bs] = logf(deg + 1.0f);
  __syncthreads();

  // -------- masked mean / max / std per feature dim (2 dims per lane) ------
  const int d0 = lane, d1 = lane + 32;
  float sum0 = 0.f, sum1 = 0.f, sq0 = 0.f, sq1 = 0.f;
  float mx0 = -3.4028234663852886e38f, mx1 = -3.4028234663852886e38f;
  for (int k = 0; k < KK; ++k) {
    const float wk = s_w[wave][k];                    // LDS broadcast
    const float f0 = s_feat[wave][k*FSTRIDE + d0];    // conflict-free column read
    const float f1 = s_feat[wave][k*FSTRIDE + d1];
    const float wf0 = wk*f0, wf1 = wk*f1;
    sum0 += wf0; sum1 += wf1;
    sq0  += wf0*wf0; sq1 += wf1*wf1;                  // adj*wf^2 == wf^2 (wf=0 if adj=0)
    if (wk != 0.0f) { mx0 = fmaxf(mx0, wf0); mx1 = fmaxf(mx1, wf1); }
  }
  const float denom = deg + EPSF;
  const float mean0 = sum0/denom, mean1 = sum1/denom;
  const float std0 = sqrtf(fmaxf(sq0/denom - mean0*mean0, 0.0f) + EPSF);
  const float std1 = sqrtf(fmaxf(sq1/denom - mean1*mean1, 0.0f) + EPSF);
  float fx0 = mx0, fx1 = mx1;
  if (deg <= 0.0f) { fx0 = 0.0f; fx1 = 0.0f; }
  const size_t base = (size_t)bs * 192;
  m_base[base +       d0] = mean0;  m_base[base +       d1] = mean1;
  m_base[base +  64 + d0] = fx0;    m_base[base +  64 + d1] = fx1;
  m_base[base + 128 + d0] = std0;   m_base[base + 128 + d1] = std1;
}

// ---------------------------------------------------------------------------
// Kernel 2: deterministic fixed-order reduction of log(deg+1) -> avg_d
// ---------------------------------------------------------------------------
__global__ __launch_bounds__(256) void gac_avg(const float* __restrict__ dlog,
                                               float* __restrict__ avg)
{
  __shared__ float red[256];
  const int tid = threadIdx.x;
  float s = 0.0f;
  for (int i = tid; i < BS; i += 256) s += dlog[i];
  red[tid] = s; __syncthreads();
  for (int o = 128; o > 0; o >>= 1) {
    if (tid < o) red[tid] += red[tid + o];
    __syncthreads();
  }
  if (tid == 0) *avg = red[0] * (1.0f / BS);
}

// ---------------------------------------------------------------------------
// Kernel 3: out[16384,128] = relu(M_full[16384,576] @ W_out + b_out)
// M_full cols: [0,192)=m, [192,384)=m*dlog/avg, [384,576)=m*avg/(dlog+eps).
// 32-row x 128-col block; each wave: two 16x16 accumulators sharing one B
// fragment per k-step; segment loops with hoisted scales (branch-free body).
// ---------------------------------------------------------------------------
__global__ __launch_bounds__(256) void gac_gemm(
    const float* __restrict__ m_base, const float* __restrict__ dlog,
    const float* __restrict__ avgp, const float* __restrict__ W_out,
    const float* __restrict__ b_out, float* __restrict__ out)
{
  __shared__ float s_m[32*193];                 // padded stride: bank-spread A reads
  __shared__ float s_s1[32], s_s2[32];
  const int tid = threadIdx.x, wave = tid >> 5, lane = tid & 31;
  const int r0 = blockIdx.x * 32;
  const float avg = *avgp;

  for (int i = tid; i < 32*192; i += 256) {
    const int r = i / 192, c = i - r*192;
    s_m[r*193 + c] = m_base[(size_t)(r0 + r)*192 + c];
  }
  if (tid < 32) {
    const float dl = dlog[r0 + tid];
    s_s1[tid] = dl / avg;
    s_s2[tid] = avg / (dl + EPSF);
  }
  __syncthreads();

  const int n0 = wave * 16;          // this wave's 16 output columns
  const int nn = lane & 15;
  const int kb = (lane >> 4) * 2;    // A/B VGPR K split: lanes0-15 K+0/1, 16-31 K+2/3
  const int Mr = lane & 15;
  const float s1a = s_s1[Mr],      s2a = s_s2[Mr];
  const float s1b = s_s1[Mr + 16], s2b = s_s2[Mr + 16];

  v8f acc0 = {0.f,0.f,0.f,0.f,0.f,0.f,0.f,0.f};
  v8f acc1 = {0.f,0.f,0.f,0.f,0.f,0.f,0.f,0.f};

  auto seg_mac = [&](int segbase, float sa, float sb) {
    for (int t = 0; t < 48; ++t) {              // 48 * K4 = one 192-col segment
      const int col = t*4 + kb;
      const int kr  = segbase + col;
      if ((t & 15) == 0 && kr + 64 < 576)
        __builtin_prefetch(&W_out[(size_t)(kr + 64) * 128 + n0 + nn], 0, 1);
      v2f bf;
      bf.x = W_out[(size_t)kr       * 128 + n0 + nn];
      bf.y = W_out[(size_t)(kr + 1) * 128 + n0 + nn];
      v2f a0, a1;
      a0.x = s_m[Mr*193 + col]            * sa;
      a0.y = s_m[Mr*193 + col + 1]        * sa;
      a1.x = s_m[(Mr + 16)*193 + col]     * sb;
      a1.y = s_m[(Mr + 16)*193 + col + 1] * sb;
      acc0 = __builtin_amdgcn_wmma_f32_16x16x4_f32(
          false, a0, false, bf, (short)0, acc0, false, false);
      acc1 = __builtin_amdgcn_wmma_f32_16x16x4_f32(
          false, a1, false, bf, (short)0, acc1, false, false);
    }
  };
  seg_mac(0,   1.0f, 1.0f);   // m
  seg_mac(192, s1a,  s1b);    // m * dlog/avg
  seg_mac(384, s2a,  s2b);    // m * avg/(dlog+eps)

  const float bias = b_out[n0 + nn];
  const int hi = (lane >> 4) * 8;               // D layout: VGPR r -> row r / r+8
  for (int r = 0; r < 8; ++r) {
    const int M = r + hi;
    out[(size_t)(r0 + M)*128      + n0 + nn] = fmaxf(acc0[r] + bias, 0.0f);
    out[(size_t)(r0 + 16 + M)*128 + n0 + nn] = fmaxf(acc1[r] + bias, 0.0f);
  }
}

// ---------------------------------------------------------------------------
extern "C" void kernel_launch(void* const* d_in, const int* in_sizes, int n_in,
                              void* d_out, int out_size, void* d_ws, size_t ws_size,
                              hipStream_t stream) {
  const float* xyz    = (const float*)d_in[0];
  const float* points = (const float*)d_in[1];
  const float* W_att  = (const float*)d_in[2];
  const float* b_att  = (const float*)d_in[3];
  const float* W_out  = (const float*)d_in[4];
  const float* b_out  = (const float*)d_in[5];
  const int*   fps    = (const int*)d_in[6];
  float* out = (float*)d_out;

  float* m_base = (float*)d_ws;                   // BS*192 floats (~12.6 MB)
  float* dlogp  = m_base + (size_t)BS*192;        // BS floats
  float* avgp   = dlogp + BS;                     // 1 float

  gac_group_att<<<BS/W1, W1*32, 0, stream>>>(xyz, points, W_att, b_att, fps,
                                             m_base, dlogp);
  gac_avg<<<1, 256, 0, stream>>>(dlogp, avgp);
  gac_gemm<<<BS/32, 256, 0, stream>>>(m_base, dlogp, avgp, W_out, b_out, out);
}